// GAT_55344948576482
// MI455X (gfx1250) — compile-verified
//
#include <hip/hip_runtime.h>
#include <hip/hip_bf16.h>

typedef float v2f __attribute__((ext_vector_type(2)));
typedef float v4f __attribute__((ext_vector_type(4)));
typedef float v8f __attribute__((ext_vector_type(8)));

#define ALPHA    0.2f
#define NEG_INF  -9e15f
#define D        128
#define KN       32      // neighbors per item
#define W1       4       // waves (items) per block in kernel 1

// ---------- wave32 reductions ----------
__device__ __forceinline__ float wave_sum(float v) {
    v += __shfl_xor(v, 16);
    v += __shfl_xor(v, 8);
    v += __shfl_xor(v, 4);
    v += __shfl_xor(v, 2);
    v += __shfl_xor(v, 1);
    return v;
}
__device__ __forceinline__ float wave_max(float v) {
    v = fmaxf(v, __shfl_xor(v, 16));
    v = fmaxf(v, __shfl_xor(v, 8));
    v = fmaxf(v, __shfl_xor(v, 4));
    v = fmaxf(v, __shfl_xor(v, 2));
    v = fmaxf(v, __shfl_xor(v, 1));
    return v;
}

// =====================================================================
// Kernel 1: fused attention + masked softmax + domain-weighted aggregate
// One wave32 per item; lane L owns dims [4L, 4L+4). Streams ent/rel once
// with non-temporal loads (read-once data; keep L2/WGP$ for reusables).
// =====================================================================
__global__ __launch_bounds__(W1 * 32)
void gat_aggregate_kernel(const float* __restrict__ item_embs,
                          const float* __restrict__ entity_embs,
                          const float* __restrict__ relations_embed,
                          const int*   __restrict__ relation_ids,
                          const int*   __restrict__ adj_mask,
                          const float* __restrict__ fc_w,
                          const float* __restrict__ fc_b,
                          const float* __restrict__ rel_dom_probs,
                          float*       __restrict__ Xout,
                          int N, int R) {
    // per-wave product stash: [wave][k][lane] float4 -> 64 KB total
    __shared__ v4f prodLds[W1][KN][32];

    const int lane = threadIdx.x & 31;
    const int wave = threadIdx.x >> 5;
    const int n = blockIdx.x * W1 + wave;
    if (n >= N) return;   // whole-wave exit; no block barriers used

    // item slice + attention weights for this lane's 4 dims
    const v4f it = *(const v4f*)(item_embs + (size_t)n * D + lane * 4);
    const v4f* fw = (const v4f*)fc_w;
    const v4f f1 = fw[lane];        // fc_w[0   :128)
    const v4f f2 = fw[32 + lane];   // fc_w[128 :256)  (relation part)
    const v4f f3 = fw[64 + lane];   // fc_w[256 :384)  (entity part)

    float eitem = it.x * f1.x + it.y * f1.y + it.z * f1.z + it.w * f1.w;
    eitem = wave_sum(eitem) + fc_b[0];

    const v4f* entp = (const v4f*)(entity_embs     + (size_t)n * KN * D);
    const v4f* relp = (const v4f*)(relations_embed + (size_t)n * KN * D);
    const int* adjp = adj_mask + (size_t)n * KN;

    float myE = NEG_INF;   // lane k will hold masked score e_k
    #pragma unroll 4
    for (int k = 0; k < KN; ++k) {
        // read-once streams -> non-temporal (TH=NT)
        const v4f e4 = __builtin_nontemporal_load(entp + k * 32 + lane);
        const v4f r4 = __builtin_nontemporal_load(relp + k * 32 + lane);

        v4f p4 = e4 * r4;              // rel*ent elementwise
        prodLds[wave][k][lane] = p4;   // stash for pass 2

        float part = r4.x * f2.x + r4.y * f2.y + r4.z * f2.z + r4.w * f2.w
                   + e4.x * f3.x + e4.y * f3.y + e4.z * f3.z + e4.w * f3.w;
        float ek = wave_sum(part) + eitem;
        ek = (ek > 0.f) ? ek : ALPHA * ek;              // LeakyReLU
        ek = (adjp[k] > 0) ? ek : NEG_INF;              // mask
        if (lane == k) myE = ek;
    }

    // masked softmax across the wave (lane k holds e_k)
    const float mx = wave_max(myE);
    const float ex = expf(myE - mx);
    const float sm = wave_sum(ex);
    const float pi = ex / sm;

    // a_total: sum over domains of rel_dom_probs[rel_id] (S == 4)
    const int id = relation_ids[(size_t)n * KN + lane];
    float atot = 0.f;
    if (id >= 0 && id < R) {
        const v4f rd = *(const v4f*)(rel_dom_probs + (size_t)id * 4);
        atot = rd.x + rd.y + rd.z + rd.w;
    }
    const float w = pi * atot;   // lane k holds combined weight w_k

    // pass 2: agg[dim] = sum_k w_k * prod[k][dim]   (LDS only)
    v4f agg = {0.f, 0.f, 0.f, 0.f};
    #pragma unroll
    for (int k = 0; k < KN; ++k) {
        const float wk = __shfl(w, k);       // readlane broadcast
        const v4f p = prodLds[wave][k][lane];
        agg += wk * p;
    }

    // residual, write X = agg + item
    const v4f xo = agg + it;
    *(v4f*)(Xout + (size_t)n * D + lane * 4) = xo;
}

// =====================================================================
// Kernel 2: out = relu(X @ out_w.T + out_b) via V_WMMA_F32_16X16X4_F32
// Block = 256 threads (8 waves), one 16-row stripe per block,
// wave w computes the 16x16 tile at columns [16w, 16w+16).
// A/B lane layout per ISA 7.12.2 (f32 16x4):
//   VGPR0: lanes0-15 K=0, lanes16-31 K=2 ; VGPR1: K=1 / K=3
// =====================================================================
__global__ __launch_bounds__(256)
void out_gemm_relu_kernel(const float* __restrict__ X,
                          const float* __restrict__ out_w,   // [D][D], row j = weights for out dim j
                          const float* __restrict__ out_b,
                          float*       __restrict__ out,
                          int N) {
    const int lane    = threadIdx.x & 31;
    const int wave    = threadIdx.x >> 5;   // col tile 0..7
    const int rowBase = blockIdx.x * 16;
    const int colBase = wave * 16;
    const int half    = lane >> 4;          // 0: K pair 0,1 ; 1: K pair 2,3
    const int l15     = lane & 15;
    const int koff    = half * 2;

    int arow = rowBase + l15;
    if (arow >= N) arow = N - 1;            // clamp (stores are bounds-checked)

    const float* aptr = X     + (size_t)arow * D + koff;                 // A[m][k]
    const float* bptr = out_w + (size_t)(colBase + l15) * D + koff;      // B[k][n] = out_w[n][k]

    v8f c = {0.f, 0.f, 0.f, 0.f, 0.f, 0.f, 0.f, 0.f};
    #pragma unroll
    for (int k0 = 0; k0 < D; k0 += 4) {
        v2f a = *(const v2f*)(aptr + k0);   // {A[m][k0+koff], A[m][k0+koff+1]}
        v2f b = *(const v2f*)(bptr + k0);   // {B[k0+koff][n], B[k0+koff+1][n]}
        c = __builtin_amdgcn_wmma_f32_16x16x4_f32(
                /*neg_a=*/false, a, /*neg_b=*/false, b,
                /*c_mod=*/(short)0, c, /*reuse_a=*/false, /*reuse_b=*/false);
    }

    const float bias = out_b[colBase + l15];
    #pragma unroll
    for (int r = 0; r < 8; ++r) {
        // C/D layout: VGPR r -> M = r (lanes 0-15) / r+8 (lanes 16-31), N = lane&15
        const int row = rowBase + r + half * 8;
        float v = c[r] + bias;
        v = (v > 0.f) ? v : 0.f;            // ReLU
        if (row < N) out[(size_t)row * D + colBase + l15] = v;
    }
}

extern "C" void kernel_launch(void* const* d_in, const int* in_sizes, int n_in,
                              void* d_out, int out_size, void* d_ws, size_t ws_size,
                              hipStream_t stream) {
    const float* item_embs       = (const float*)d_in[0];
    const float* entity_embs     = (const float*)d_in[1];
    const float* relations_embed = (const float*)d_in[2];
    const int*   relation_ids    = (const int*)  d_in[3];
    const int*   adj_mask        = (const int*)  d_in[4];
    const float* fc_w            = (const float*)d_in[5];
    const float* fc_b            = (const float*)d_in[6];
    const float* out_w           = (const float*)d_in[7];
    const float* out_b           = (const float*)d_in[8];
    const float* rel_dom_probs   = (const float*)d_in[9];

    const int N = in_sizes[0] / D;          // 20000
    const int R = in_sizes[9] / 4;          // S == 4 domains per relation

    float* X = (float*)d_ws;                // N*D floats intermediate

    // Kernel 1: streaming fused attention/aggregation (bandwidth-bound part)
    {
        dim3 grid((N + W1 - 1) / W1);
        dim3 block(W1 * 32);
        gat_aggregate_kernel<<<grid, block, 0, stream>>>(
            item_embs, entity_embs, relations_embed, relation_ids, adj_mask,
            fc_w, fc_b, rel_dom_probs, X, N, R);
    }

    // Kernel 2: fp32 WMMA output projection + bias + ReLU
    {
        dim3 grid((N + 15) / 16);
        dim3 block(256);
        out_gemm_relu_kernel<<<grid, block, 0, stream>>>(
            X, out_w, out_b, (float*)d_out, N);
    }
}